// DotProductHead_10539849744621
// MI455X (gfx1250) — compile-verified
//
#include <hip/hip_runtime.h>

// CDNA5 (gfx1250) wave32 WMMA types
typedef __attribute__((ext_vector_type(2))) float v2f;  // A/B fragment of 16x16x4 f32 (2 VGPRs/lane)
typedef __attribute__((ext_vector_type(8))) float v8f;  // C/D 16x16 f32 accumulator (8 VGPRs/lane)

// One wave32 computes 16 edge dot-products as diag(S * T^T) via 32 chained
// v_wmma_f32_16x16x4_f32 accumulations over K=128 (full fp32 precision).
//
// A (16x4 f32): lane L holds row M=L&15, elements K = 2*(L>>4) .. 2*(L>>4)+1  -> float2 load
// B (4x16 f32): lane L holds col N=L&15, rows  K = 2*(L>>4) .. 2*(L>>4)+1     -> float2 load
// => both fragments are contiguous 8B loads at  rowPtr + k + 2*(lane>>4).
__global__ __launch_bounds__(256) void DotProductHead_wmma_kernel(
    const float* __restrict__ emb,     // [100000, 128] fp32
    const int*   __restrict__ eidx,    // [2, nEdges]
    float*       __restrict__ out,     // [nEdges]
    int nEdges, int nGroups)
{
    const int lane = threadIdx.x & 31;
    const int wave = threadIdx.x >> 5;
    const int group = blockIdx.x * 8 + wave;     // 8 waves per 256-thread block
    if (group >= nGroups) return;                // wave-uniform exit (EXEC stays all-1 inside)

    const int m  = lane & 15;                    // edge within group == matrix row/col
    const int hi = lane >> 4;                    // lane half selects K pair {0,1} vs {2,3}

    int e = group * 16 + m;
    if (e >= nEdges) e = nEdges - 1;             // clamp: keep every lane active for WMMA

    const long long srcRow = (long long)eidx[e];            // edge_index[0][e]
    const long long tgtRow = (long long)eidx[nEdges + e];   // edge_index[1][e]

    const float* sp = emb + srcRow * 128 + 2 * hi;
    const float* tp = emb + tgtRow * 128 + 2 * hi;

    v8f c = {};
#pragma unroll
    for (int k = 0; k < 128; k += 4) {
        v2f a = *(const v2f*)(sp + k);   // global_load_b64, L2-resident gather
        v2f b = *(const v2f*)(tp + k);
        // 8 args: (neg_a, A, neg_b, B, c_mod, C, reuse_a, reuse_b)
        c = __builtin_amdgcn_wmma_f32_16x16x4_f32(
                false, a, false, b, (short)0, c, false, false);
    }

    // diag(D): (m,m) for m<8 lives in lane m, c[m]; for m>=8 in lane m+16, c[m-8].
    // Owner lanes: 0..7 and 24..31; component index = lane & 7 in both cases.
    const int idx = lane & 7;
    float v = c[0];
    v = (idx == 1) ? c[1] : v;
    v = (idx == 2) ? c[2] : v;
    v = (idx == 3) ? c[3] : v;
    v = (idx == 4) ? c[4] : v;
    v = (idx == 5) ? c[5] : v;
    v = (idx == 6) ? c[6] : v;
    v = (idx == 7) ? c[7] : v;
    v *= (1.0f / 128.0f);                        // mean over embedding dim

    const bool owner = (lane < 8) || (lane >= 24);
    const int eo = group * 16 + m;
    if (owner && eo < nEdges) out[eo] = v;
}

extern "C" void kernel_launch(void* const* d_in, const int* in_sizes, int n_in,
                              void* d_out, int out_size, void* d_ws, size_t ws_size,
                              hipStream_t stream) {
    const float* emb  = (const float*)d_in[0];   // node_embeddings fp32 [100000*128]
    const int*   eidx = (const int*)d_in[1];     // edge_index int [2*600000]
    float*       out  = (float*)d_out;           // fp32 [600000]

    const int nEdges  = in_sizes[1] / 2;         // 600000
    const int nGroups = (nEdges + 15) / 16;      // 37500 (exact, no tail)
    const int wavesPerBlock = 8;                 // 256 threads = 8 wave32
    const int blocks = (nGroups + wavesPerBlock - 1) / wavesPerBlock;

    DotProductHead_wmma_kernel<<<blocks, 256, 0, stream>>>(emb, eidx, out, nEdges, nGroups);
}